// SimGNN_68865505624176
// MI455X (gfx1250) — compile-verified
//
#include <hip/hip_runtime.h>
#include <math.h>

typedef __attribute__((ext_vector_type(2))) float v2f;
typedef __attribute__((ext_vector_type(8))) float v8f;

#define DEV static __device__ __forceinline__

constexpr int NTOT   = 65536;   // B*N total nodes per side
constexpr int NEDGE  = 524288;  // E edges per side
constexpr int NGRAPH = 128;     // B graphs
constexpr int NNODE  = 512;     // N nodes per graph
constexpr int FDIM   = 64;      // F3

DEV v8f wmma4_f32(v2f a, v2f b, v8f c) {
  // V_WMMA_F32_16X16X4_F32 : exact f32 16x16 tile MAC, K=4 per issue
  return __builtin_amdgcn_wmma_f32_16x16x4_f32(false, a, false, b, (short)0, c,
                                               false, false);
}

DEV float sigm(float x) { return 1.f / (1.f + expf(-x)); }

DEV void atomicMaxF(float* addr, float v) {
  if (v >= 0.f) atomicMax((int*)addr, __float_as_int(v));
  else          atomicMin((unsigned int*)addr, __float_as_uint(v));
}
DEV void atomicMinF(float* addr, float v) {
  if (v >= 0.f) atomicMin((int*)addr, __float_as_int(v));
  else          atomicMax((unsigned int*)addr, __float_as_uint(v));
}

// ---------------- generic small kernels ----------------

__global__ void fill_f32(float* p, float v, int n) {
  int i = blockIdx.x * blockDim.x + threadIdx.x;
  if (i < n) p[i] = v;
}

__global__ void deg_edges(const int* __restrict__ dst, float* __restrict__ deg, int e) {
  int i = blockIdx.x * blockDim.x + threadIdx.x;
  if (i < e) atomicAdd(&deg[dst[i]], 1.0f);
}

__global__ void to_rsqrt(float* p, int n) {
  int i = blockIdx.x * blockDim.x + threadIdx.x;
  if (i < n) p[i] = rsqrtf(p[i]);
}

// ---------------- WMMA GEMM: C[MxN] = A[MxK] @ B[KxN], row-major ----------------
// One wave32 owns one 16x16 C tile; K stepped by 4 (V_WMMA_F32_16X16X4_F32).
// A layout: lane(l,h): row tm*16+l, ks {kb+2h, kb+2h+1}. B: col tn*16+l, same ks.
// C layout: vgpr r -> row r + 8h, col l.

__global__ void gemm_wmma_f32(const float* __restrict__ A, const float* __restrict__ B,
                              float* __restrict__ C, int M, int K, int N) {
  int wave = (int)((blockIdx.x * blockDim.x + threadIdx.x) >> 5);
  int lane = threadIdx.x & 31;
  int tilesN = N >> 4;
  int totalTiles = (M >> 4) * tilesN;
  if (wave >= totalTiles) return;           // wave-uniform: EXEC stays all-ones
  int tm = wave / tilesN;
  int tn = wave - tm * tilesN;
  int h = lane >> 4, l = lane & 15;
  const float* arow = A + (size_t)(tm * 16 + l) * K;
  const float* bcol = B + (tn * 16 + l);
  v8f acc = {0.f, 0.f, 0.f, 0.f, 0.f, 0.f, 0.f, 0.f};
  for (int kb = 0; kb < K; kb += 4) {
    int k0 = kb + 2 * h;
    v2f a, b;
    a.x = arow[k0];
    a.y = arow[k0 + 1];
    b.x = bcol[(size_t)k0 * N];
    b.y = bcol[(size_t)(k0 + 1) * N];
    acc = wmma4_f32(a, b, acc);
  }
  float* crow = C + (size_t)(tm * 16 + 8 * h) * N + (size_t)(tn * 16 + l);
#pragma unroll
  for (int r = 0; r < 8; ++r) crow[(size_t)r * N] = acc[r];
}

// ---------------- GCN aggregation ----------------

__global__ void self_term(const float* __restrict__ H, const float* __restrict__ dinv,
                          float* __restrict__ outb, int n, int F) {
  int i = blockIdx.x * blockDim.x + threadIdx.x;
  if (i >= n) return;
  int node = i / F;
  float di = dinv[node];
  outb[i] = H[i] * di * di;
}

__global__ void edge_agg(const int* __restrict__ src, const int* __restrict__ dst,
                         const float* __restrict__ H, const float* __restrict__ dinv,
                         float* __restrict__ outb, int e, int F) {
  long long idx = (long long)blockIdx.x * blockDim.x + threadIdx.x;
  long long tot = (long long)e * F;
  if (idx >= tot) return;
  int ed = (int)(idx / F);
  int fc = (int)(idx - (long long)ed * F);
  int s = src[ed], d = dst[ed];
  float nrm = dinv[s] * dinv[d];
  atomicAdd(&outb[(size_t)d * F + fc], H[(size_t)s * F + fc] * nrm);
}

__global__ void bias_act(const float* __restrict__ in, const float* __restrict__ bias,
                         float* __restrict__ outb, int n, int F, int act) {
  int i = blockIdx.x * blockDim.x + threadIdx.x;
  if (i >= n) return;
  float v = in[i] + bias[i % F];
  if (act) v = fmaxf(v, 0.f);
  outb[i] = v;
}

// ---------------- similarity matrix (WMMA) : pass1 min/max, pass2 histogram ----------------
// S_g = F1_g (512x64) @ F2_g^T ; B element (k,n) = F2_g[n][k] (row-major source).

DEV v8f sim_tile(const float* __restrict__ f1, const float* __restrict__ f2,
                 int g, int tm, int tn, int lane) {
  int h = lane >> 4, l = lane & 15;
  const float* arow = f1 + ((size_t)g * NNODE + (size_t)(tm * 16 + l)) * FDIM;
  const float* brow = f2 + ((size_t)g * NNODE + (size_t)(tn * 16 + l)) * FDIM;
  v8f acc = {0.f, 0.f, 0.f, 0.f, 0.f, 0.f, 0.f, 0.f};
  for (int kb = 0; kb < FDIM; kb += 4) {
    int k0 = kb + 2 * h;
    v2f a, b;
    a.x = arow[k0];
    a.y = arow[k0 + 1];
    b.x = brow[k0];
    b.y = brow[k0 + 1];
    acc = wmma4_f32(a, b, acc);
  }
  return acc;
}

__global__ void sim_minmax(const float* __restrict__ f1, const float* __restrict__ f2,
                           float* __restrict__ smin, float* __restrict__ smax) {
  int g = blockIdx.z, tm = blockIdx.y, tn = blockIdx.x;
  int lane = threadIdx.x;
  v8f acc = sim_tile(f1, f2, g, tm, tn, lane);
  float mn = acc[0], mx = acc[0];
#pragma unroll
  for (int r = 1; r < 8; ++r) {
    mn = fminf(mn, acc[r]);
    mx = fmaxf(mx, acc[r]);
  }
  for (int off = 16; off > 0; off >>= 1) {
    mn = fminf(mn, __shfl_xor(mn, off, 32));
    mx = fmaxf(mx, __shfl_xor(mx, off, 32));
  }
  if (lane == 0) {
    atomicMinF(&smin[g], mn);
    atomicMaxF(&smax[g], mx);
  }
}

__global__ void sim_hist(const float* __restrict__ f1, const float* __restrict__ f2,
                         const float* __restrict__ smin, const float* __restrict__ smax,
                         int* __restrict__ hcnt) {
  __shared__ int lh[16];
  int g = blockIdx.z, tm = blockIdx.y, tn = blockIdx.x;
  int lane = threadIdx.x;
  if (lane < 16) lh[lane] = 0;
  __syncthreads();
  v8f acc = sim_tile(f1, f2, g, tm, tn, lane);
  // sigmoid is monotone: vmin/vmax = sigmoid(smin/smax)
  float vmin = sigm(smin[g]);
  float vmax = sigm(smax[g]);
  float w = (vmax - vmin) * (1.f / 16.f);
  float safe = (w > 0.f) ? w : 1.f;
#pragma unroll
  for (int r = 0; r < 8; ++r) {
    float v = sigm(acc[r]);
    int idx = (int)floorf((v - vmin) / safe);
    idx = idx < 0 ? 0 : (idx > 15 ? 15 : idx);
    atomicAdd(&lh[idx], 1);
  }
  __syncthreads();
  if (lane < 16) atomicAdd(&hcnt[g * 16 + lane], lh[lane]);
}

// ---------------- attention pooling ----------------

__global__ void attn_ctx(const float* __restrict__ f, const float* __restrict__ Watt,
                         float* __restrict__ ctx) {
  int g = blockIdx.x, t = threadIdx.x;  // 64 threads
  __shared__ float mean[FDIM];
  const float* base = f + (size_t)g * NNODE * FDIM;
  float s = 0.f;
  for (int n = 0; n < NNODE; ++n) s += base[(size_t)n * FDIM + t];
  mean[t] = s * (1.f / (float)NNODE);
  __syncthreads();
  float c = 0.f;
  for (int i = 0; i < FDIM; ++i) c += mean[i] * Watt[i * FDIM + t];
  ctx[g * FDIM + t] = tanhf(c);
}

__global__ void attn_coef(const float* __restrict__ f, const float* __restrict__ ctx,
                          float* __restrict__ coef) {
  int i = blockIdx.x * blockDim.x + threadIdx.x;
  if (i >= NTOT) return;
  int g = i >> 9;  // / NNODE
  const float* fr = f + (size_t)i * FDIM;
  const float* c = ctx + g * FDIM;
  float s = 0.f;
  for (int d = 0; d < FDIM; ++d) s += fr[d] * c[d];
  coef[i] = sigm(s);
}

__global__ void attn_pool(const float* __restrict__ f, const float* __restrict__ coef,
                          float* __restrict__ p) {
  int g = blockIdx.x, t = threadIdx.x;  // 64 threads
  const float* base = f + (size_t)g * NNODE * FDIM;
  const float* cf = coef + (size_t)g * NNODE;
  float s = 0.f;
  for (int n = 0; n < NNODE; ++n) s += base[(size_t)n * FDIM + t] * cf[n];
  p[g * FDIM + t] = s;
}

// ---------------- NTN + histogram feature ----------------

__global__ void tensor_feat(const float* __restrict__ p1, const float* __restrict__ p2,
                            const float* __restrict__ Wt, const float* __restrict__ Wtb,
                            const float* __restrict__ tb, const int* __restrict__ hcnt,
                            float* __restrict__ feat) {
  int g = blockIdx.x, k = threadIdx.x;  // 16 threads
  const float* a = p1 + g * FDIM;
  const float* b = p2 + g * FDIM;
  float s = 0.f;
  for (int i = 0; i < FDIM; ++i) {
    float t = 0.f;
    for (int j = 0; j < FDIM; ++j) t += Wt[(size_t)(i * FDIM + j) * 16 + k] * b[j];
    s += a[i] * t;
  }
  float blk = 0.f;
  for (int c = 0; c < FDIM; ++c)
    blk += a[c] * Wtb[k * 128 + c] + b[c] * Wtb[k * 128 + 64 + c];
  float v = fmaxf(s + blk + tb[k], 0.f);
  feat[g * 32 + k] = v;
  // normalized histogram (total count is exactly N*N)
  feat[g * 32 + 16 + k] = (float)hcnt[g * 16 + k] * (1.f / (float)(NNODE * NNODE));
}

__global__ void final_mlp(const float* __restrict__ feat, const float* __restrict__ W1,
                          const float* __restrict__ b1, const float* __restrict__ Wsc,
                          const float* __restrict__ bs, float* __restrict__ out) {
  int b = blockIdx.x * blockDim.x + threadIdx.x;
  if (b >= NGRAPH) return;
  const float* fr = feat + b * 32;
  float acc = 0.f;
  for (int j = 0; j < 32; ++j) {
    float h = b1[j];
    for (int c = 0; c < 32; ++c) h += fr[c] * W1[c * 32 + j];
    h = fmaxf(h, 0.f);
    acc += h * Wsc[j];
  }
  out[b] = sigm(acc + bs[0]);
}

// ---------------- host launcher ----------------

extern "C" void kernel_launch(void* const* d_in, const int* in_sizes, int n_in,
                              void* d_out, int out_size, void* d_ws, size_t ws_size,
                              hipStream_t stream) {
  (void)in_sizes; (void)n_in; (void)out_size; (void)ws_size;
  const float* x1  = (const float*)d_in[0];
  const float* x2  = (const float*)d_in[1];
  const int*   ei1 = (const int*)d_in[2];
  const int*   ei2 = (const int*)d_in[3];
  const float* Wc1 = (const float*)d_in[6];
  const float* bc1 = (const float*)d_in[7];
  const float* Wc2 = (const float*)d_in[8];
  const float* bc2 = (const float*)d_in[9];
  const float* Wc3 = (const float*)d_in[10];
  const float* bc3 = (const float*)d_in[11];
  const float* Watt = (const float*)d_in[12];
  const float* Wt   = (const float*)d_in[13];
  const float* Wtb  = (const float*)d_in[14];
  const float* tb   = (const float*)d_in[15];
  const float* W1   = (const float*)d_in[16];
  const float* b1   = (const float*)d_in[17];
  const float* Wsc  = (const float*)d_in[18];
  const float* bs   = (const float*)d_in[19];
  float* out = (float*)d_out;

  float* ws = (float*)d_ws;
  size_t o = 0;
  float* bufA = ws + o; o += (size_t)NTOT * 128;   // GEMM output / agg input
  float* bufB = ws + o; o += (size_t)NTOT * 128;   // agg output / next activations
  float* f1   = ws + o; o += (size_t)NTOT * FDIM;
  float* f2   = ws + o; o += (size_t)NTOT * FDIM;
  float* dinv = ws + o; o += NTOT;
  float* coef = ws + o; o += NTOT;
  float* smin = ws + o; o += NGRAPH;
  float* smax = ws + o; o += NGRAPH;
  int*   hcnt = (int*)(ws + o); o += NGRAPH * 16;
  float* ctx  = ws + o; o += NGRAPH * FDIM;
  float* p1   = ws + o; o += NGRAPH * FDIM;
  float* p2   = ws + o; o += NGRAPH * FDIM;
  float* feat = ws + o; o += NGRAPH * 32;

  auto gemm = [&](const float* A, const float* Bm, float* C, int M, int K, int N) {
    int tiles = (M / 16) * (N / 16);
    int blocks = (tiles + 7) / 8;  // 8 waves (256 threads) per block
    gemm_wmma_f32<<<blocks, 256, 0, stream>>>(A, Bm, C, M, K, N);
  };

  auto layer = [&](const float* in, const float* W, const float* bias, const int* ei,
                   int K, int N, int act, float* outbuf) {
    gemm(in, W, bufA, NTOT, K, N);  // H = X @ W
    int n = NTOT * N;
    self_term<<<(n + 255) / 256, 256, 0, stream>>>(bufA, dinv, bufB, n, N);
    long long tot = (long long)NEDGE * N;
    edge_agg<<<(unsigned)((tot + 255) / 256), 256, 0, stream>>>(ei, ei + NEDGE, bufA,
                                                               dinv, bufB, NEDGE, N);
    bias_act<<<(n + 255) / 256, 256, 0, stream>>>(bufB, bias, outbuf, n, N, act);
  };

  auto side = [&](const float* x, const int* ei, float* fout) {
    fill_f32<<<(NTOT + 255) / 256, 256, 0, stream>>>(dinv, 1.0f, NTOT);  // deg = 1 (self)
    deg_edges<<<(NEDGE + 255) / 256, 256, 0, stream>>>(ei + NEDGE, dinv, NEDGE);
    to_rsqrt<<<(NTOT + 255) / 256, 256, 0, stream>>>(dinv, NTOT);        // dinv = deg^-1/2
    layer(x,    Wc1, bc1, ei,  32, 128, 1, bufB);
    layer(bufB, Wc2, bc2, ei, 128,  96, 1, bufB);
    layer(bufB, Wc3, bc3, ei,  96,  64, 0, fout);
  };

  side(x1, ei1, f1);
  side(x2, ei2, f2);

  // histogram of sigmoid(f1 f2^T), recomputed twice to avoid 134MB spill
  fill_f32<<<1, 256, 0, stream>>>(smin, INFINITY, NGRAPH);
  fill_f32<<<1, 256, 0, stream>>>(smax, -INFINITY, NGRAPH);
  fill_f32<<<(NGRAPH * 16 + 255) / 256, 256, 0, stream>>>((float*)hcnt, 0.f, NGRAPH * 16);
  dim3 sg(NNODE / 16, NNODE / 16, NGRAPH);
  sim_minmax<<<sg, 32, 0, stream>>>(f1, f2, smin, smax);
  sim_hist<<<sg, 32, 0, stream>>>(f1, f2, smin, smax, hcnt);

  // attention pooling per side
  attn_ctx<<<NGRAPH, FDIM, 0, stream>>>(f1, Watt, ctx);
  attn_coef<<<(NTOT + 255) / 256, 256, 0, stream>>>(f1, ctx, coef);
  attn_pool<<<NGRAPH, FDIM, 0, stream>>>(f1, coef, p1);
  attn_ctx<<<NGRAPH, FDIM, 0, stream>>>(f2, Watt, ctx);
  attn_coef<<<(NTOT + 255) / 256, 256, 0, stream>>>(f2, ctx, coef);
  attn_pool<<<NGRAPH, FDIM, 0, stream>>>(f2, coef, p2);

  // tensor network + final scorer
  tensor_feat<<<NGRAPH, 16, 0, stream>>>(p1, p2, Wt, Wtb, tb, hcnt, feat);
  final_mlp<<<1, NGRAPH, 0, stream>>>(feat, W1, b1, Wsc, bs, out);
}